// HeteroRGCNLayer_5995774345996
// MI455X (gfx1250) — compile-verified
//
#include <hip/hip_runtime.h>

typedef __attribute__((ext_vector_type(2))) float v2f;
typedef __attribute__((ext_vector_type(4))) float v4f;
typedef __attribute__((ext_vector_type(8))) float v8f;

#define N_NODES 50000
#define N_EDGES 800000
#define IN_SIZE 256
#define OUT_SIZE 256

// ---------------------------------------------------------------------------
// Kernel 0: zero the output accumulator and the degree array
// ---------------------------------------------------------------------------
__global__ void zero_kernel(float* __restrict__ out, float* __restrict__ deg) {
    size_t tid = (size_t)blockIdx.x * blockDim.x + threadIdx.x;
    size_t stride = (size_t)gridDim.x * blockDim.x;
    size_t total = (size_t)N_NODES * OUT_SIZE;
    for (size_t j = tid; j < total; j += stride) out[j] = 0.0f;
    for (size_t j = tid; j < N_NODES; j += stride) deg[j] = 0.0f;
}

// ---------------------------------------------------------------------------
// Kernel 1: Wh = x @ W + b   via V_WMMA_F32_16X16X4_F32 (fp32 tensor path)
// Block = 256 threads = 8 waves. Block owns a 16-row strip (M tile);
// each wave owns two 16-wide N tiles -> 8 waves cover all 256 columns.
// 50000 rows = 3125 exact 16-row tiles (no tail).
// ---------------------------------------------------------------------------
__global__ __launch_bounds__(256)
void gemm_wmma_kernel(const float* __restrict__ x, const float* __restrict__ W,
                      const float* __restrict__ bias, float* __restrict__ Wh) {
    const int lane = threadIdx.x & 31;   // wave32
    const int wave = threadIdx.x >> 5;   // 0..7
    const int col  = lane & 15;
    const int hf   = lane >> 4;          // 0 or 1 (lane half)
    const int mbase = blockIdx.x * 16;

    const int n0 = wave * 32;            // first N tile
    const int n1 = n0 + 16;              // second N tile

    v8f acc0 = {};
    v8f acc1 = {};

    // A-fragment row for this lane (16x4 f32 A layout: M = lane&15)
    const float* xrow = x + (size_t)(mbase + col) * IN_SIZE;

    for (int k0 = 0; k0 < IN_SIZE; k0 += 4) {
        // A 16x4: lanes 0-15 hold K={k0,k0+1}, lanes 16-31 hold K={k0+2,k0+3}
        const int ka = k0 + 2 * hf;
        v2f a = *(const v2f*)(xrow + ka);

        // B 4x16: K split across lane halves, consecutive K in the 2 VGPRs
        v2f b0, b1;
        b0.x = W[(size_t)ka       * OUT_SIZE + n0 + col];
        b0.y = W[(size_t)(ka + 1) * OUT_SIZE + n0 + col];
        b1.x = W[(size_t)ka       * OUT_SIZE + n1 + col];
        b1.y = W[(size_t)(ka + 1) * OUT_SIZE + n1 + col];

        // 8 args: (neg_a, A, neg_b, B, c_mod, C, reuse_a, reuse_b)
        acc0 = __builtin_amdgcn_wmma_f32_16x16x4_f32(false, a, false, b0,
                                                     (short)0, acc0, false, false);
        acc1 = __builtin_amdgcn_wmma_f32_16x16x4_f32(false, a, false, b1,
                                                     (short)0, acc1, false, false);
    }

    const float bias0 = bias[n0 + col];
    const float bias1 = bias[n1 + col];

    // C/D layout: VGPR r -> row (r + 8*half), col = lane&15
    #pragma unroll
    for (int r = 0; r < 8; ++r) {
        const size_t row = (size_t)(mbase + r + 8 * hf);
        Wh[row * OUT_SIZE + n0 + col] = acc0[r] + bias0;
        Wh[row * OUT_SIZE + n1 + col] = acc1[r] + bias1;
    }
}

// ---------------------------------------------------------------------------
// Kernel 2: per-edge gather + L2-side atomic scatter-add (+ degree count)
// One wave32 per edge; each lane moves two float4 (coalesced b128 loads)
// and issues 8 global_atomic_add_f32 (no-return -> STOREcnt only).
// Wh (51 MB) and out (51 MB) are L2-resident (192 MB), so the random
// traffic never leaves the chip.
// ---------------------------------------------------------------------------
__global__ __launch_bounds__(256)
void scatter_mean_kernel(const float* __restrict__ Wh,
                         const int* __restrict__ esrc,
                         const int* __restrict__ edst,
                         float* __restrict__ out, float* __restrict__ deg) {
    const int lane = threadIdx.x & 31;
    const int wave = threadIdx.x >> 5;
    const int e = blockIdx.x * 8 + wave;
    if (e >= N_EDGES) return;

    const int src = esrc[e];
    const int dst = edst[e];

    const v4f* s4 = (const v4f*)(Wh + (size_t)src * OUT_SIZE);
    float* drow = out + (size_t)dst * OUT_SIZE;

    #pragma unroll
    for (int j = 0; j < 2; ++j) {
        const int q = lane + 32 * j;     // float4 index within the row
        v4f v = s4[q];
        const int base = q * 4;
        atomicAdd(&drow[base + 0], v.x);
        atomicAdd(&drow[base + 1], v.y);
        atomicAdd(&drow[base + 2], v.z);
        atomicAdd(&drow[base + 3], v.w);
    }
    if (lane == 0) atomicAdd(&deg[dst], 1.0f);
}

// ---------------------------------------------------------------------------
// Kernel 3: h = sum / max(deg, 1)  in place, float4 granularity
// ---------------------------------------------------------------------------
__global__ __launch_bounds__(256)
void finalize_kernel(float* __restrict__ out, const float* __restrict__ deg) {
    const size_t i = (size_t)blockIdx.x * blockDim.x + threadIdx.x;  // one float4
    const size_t total = (size_t)N_NODES * (OUT_SIZE / 4);
    if (i >= total) return;
    const int node = (int)(i / (OUT_SIZE / 4));
    const float inv = 1.0f / fmaxf(deg[node], 1.0f);
    v4f* o4 = (v4f*)out;
    v4f v = o4[i];
    v.x *= inv; v.y *= inv; v.z *= inv; v.w *= inv;
    o4[i] = v;
}

// ---------------------------------------------------------------------------
extern "C" void kernel_launch(void* const* d_in, const int* in_sizes, int n_in,
                              void* d_out, int out_size, void* d_ws, size_t ws_size,
                              hipStream_t stream) {
    const float* x    = (const float*)d_in[0];   // [50000, 256]
    const float* W    = (const float*)d_in[1];   // [256, 256]
    const float* bias = (const float*)d_in[2];   // [256]
    const int*   esrc = (const int*)d_in[3];     // [800000]
    const int*   edst = (const int*)d_in[4];     // [800000]

    float* out = (float*)d_out;                          // [50000, 256]
    float* Wh  = (float*)d_ws;                           // 50000*256 floats
    float* deg = Wh + (size_t)N_NODES * OUT_SIZE;        // 50000 floats

    zero_kernel<<<2048, 256, 0, stream>>>(out, deg);

    gemm_wmma_kernel<<<N_NODES / 16, 256, 0, stream>>>(x, W, bias, Wh);

    scatter_mean_kernel<<<(N_EDGES + 7) / 8, 256, 0, stream>>>(Wh, esrc, edst, out, deg);

    const int n4 = N_NODES * (OUT_SIZE / 4);
    finalize_kernel<<<(n4 + 255) / 256, 256, 0, stream>>>(out, deg);
}